// PointGridEncoder_59322088292455
// MI455X (gfx1250) — compile-verified
//
#include <hip/hip_runtime.h>

// ---------------------------------------------------------------------------
// PointGridEncoder for MI455X (gfx1250, wave32).
//
// fp32 reference -> V_WMMA_F32_16X16X4_F32 everywhere.
//   * feat = h @ W2 : 16x64 @ 64x64 per wave (2 points/wave)
//   * 3D convs as implicit GEMM: wave = full w-row (4 M-tiles of 16) x 64 cout,
//     K = 27 taps x 64 cin. Weights pre-packed into the per-lane WMMA B layout
//     and staged per kd-plane in LDS (144 KB) shared by the block's 4 waves.
//
// WMMA fragment layout (CDNA5 ISA 7.12.2, wave32):
//   A (16x4 f32, 2 VGPRs): lanes 0-15 -> row M=lane,    K={0,1}
//                          lanes 16-31 -> row M=lane-16, K={2,3}
//   B (4x16 f32, 2 VGPRs): lanes 0-15 -> col N=lane,    K={0,1}
//                          lanes 16-31 -> col N=lane-16, K={2,3}
//   C/D (16x16 f32, 8 VGPRs): VGPR v: lanes 0-15 -> M=v, N=lane
//                                     lanes 16-31 -> M=v+8, N=lane-16
// ---------------------------------------------------------------------------

typedef __attribute__((ext_vector_type(2))) float v2f;
typedef __attribute__((ext_vector_type(8))) float v8f;

#define Gdim 64
#define Lch 64
#define KNN 8
#define NPTS 150000
#define NEG_SLOPE 0.01f

// packed-B sizes
#define TAP_PACK (16 * 4 * 64)          // floats per tap: [kstep][ntile][lane*2]
#define CONV_PACK (27 * TAP_PACK)       // 110592 floats per conv layer
#define PLANE_PACK (9 * TAP_PACK)       // 36864 floats per kd-plane (144 KB)

__device__ __forceinline__ float lrelu(float x) {
    return x >= 0.0f ? x : NEG_SLOPE * x;
}

// --------------------------- utility kernels -------------------------------

__global__ void zero_f32(float* __restrict__ p, int n) {
    int i = blockIdx.x * blockDim.x + threadIdx.x;
    if (i < n) p[i] = 0.0f;
}

// Cw [O=64][I=64][27] -> wPack[tap][kstep=16][ntile=4][lane*2+sub]
//   cin  = 4*kstep + 2*(lane>=16) + sub, cout = ntile*16 + (lane&15)
__global__ void pack_conv_w(const float* __restrict__ w, float* __restrict__ wPack) {
    int e = blockIdx.x * blockDim.x + threadIdx.x;
    if (e >= CONV_PACK) return;
    int sub  = e & 1;
    int lane = (e >> 1) & 31;
    int n    = (e >> 6) & 3;
    int t    = (e >> 8) & 15;
    int tap  = e >> 12;
    int cin  = 4 * t + ((lane >> 4) << 1) + sub;
    int cout = n * 16 + (lane & 15);
    wPack[e] = w[(cout * 64 + cin) * 27 + tap];
}

// W2 [j=64][ch=64] -> w2Pack[kstep=16][ntile=4][lane*2+sub]
__global__ void pack_w2(const float* __restrict__ w2, float* __restrict__ w2Pack) {
    int e = blockIdx.x * blockDim.x + threadIdx.x;
    if (e >= TAP_PACK) return;
    int sub  = e & 1;
    int lane = (e >> 1) & 31;
    int n    = (e >> 6) & 3;
    int t    = e >> 8;
    int j    = 4 * t + ((lane >> 4) << 1) + sub;
    int ch   = n * 16 + (lane & 15);
    w2Pack[e] = w2[j * 64 + ch];
}

// grid = (cnt > 0) ? sums / cnt : 0   (in place on sums)
__global__ void finalize_grid(float* __restrict__ sums, const float* __restrict__ cnt) {
    int i = blockIdx.x * blockDim.x + threadIdx.x;
    if (i >= Gdim * Gdim * Gdim * Lch) return;
    float c = cnt[i >> 6];
    sums[i] = (c > 0.0f) ? sums[i] / c : 0.0f;
}

// ----------------------- point MLP + scatter (WMMA) ------------------------
// One wave handles 2 points. A rows: M = point*8 + neighbor.
__global__ void __launch_bounds__(128)
point_encode_scatter(const float* __restrict__ pos,     // [NP][8][3]
                     const int*   __restrict__ pts,     // [NP][3]
                     const float* __restrict__ W1,      // [3][64]
                     const float* __restrict__ b1,      // [64]
                     const float* __restrict__ w2Pack,  // packed B fragments
                     const float* __restrict__ b2,      // [64]
                     const float* __restrict__ Wk1,     // [8][8]
                     const float* __restrict__ bk1,     // [8]
                     const float* __restrict__ Wk2,     // [8]
                     const float* __restrict__ bk2,     // [1]
                     float* __restrict__ sums,          // [G^3][64]
                     float* __restrict__ cnt)           // [G^3]
{
    const int lane = threadIdx.x & 31;
    const int wave = (blockIdx.x * blockDim.x + threadIdx.x) >> 5;
    const int p0 = wave * 2;
    if (p0 >= NPTS) return;
    const bool p1ok = (p0 + 1) < NPTS;
    const int p1 = p1ok ? (p0 + 1) : p0;

    const int M     = lane & 15;          // A row owned by this lane
    const int ksub  = (lane >> 4) << 1;   // K sub-slot of this lane half (0 or 2)
    const int nlane = lane & 15;

    const int pm = (M < 8) ? p0 : p1;
    const int km = M & 7;
    const float px = pos[(pm * KNN + km) * 3 + 0];
    const float py = pos[(pm * KNN + km) * 3 + 1];
    const float pz = pos[(pm * KNN + km) * 3 + 2];

    v8f acc[4] = {};
    const float* w2p = w2Pack + lane * 2;

#pragma unroll 4
    for (int t = 0; t < 16; ++t) {
        const int j = 4 * t + ksub;
        // hidden layer on the fly for this lane's two K slots
        v2f w1x = *reinterpret_cast<const v2f*>(W1 + j);
        v2f w1y = *reinterpret_cast<const v2f*>(W1 + 64 + j);
        v2f w1z = *reinterpret_cast<const v2f*>(W1 + 128 + j);
        v2f b1v = *reinterpret_cast<const v2f*>(b1 + j);
        v2f a;
        a.x = lrelu(px * w1x.x + py * w1y.x + pz * w1z.x + b1v.x);
        a.y = lrelu(px * w1x.y + py * w1y.y + pz * w1z.y + b1v.y);
#pragma unroll
        for (int n = 0; n < 4; ++n) {
            v2f b = *reinterpret_cast<const v2f*>(w2p + (t * 4 + n) * 64);
            acc[n] = __builtin_amdgcn_wmma_f32_16x16x4_f32(
                false, a, false, b, (short)0, acc[n], false, false);
        }
    }

    // acc[n][v]: lanes 0-15 -> point0 neighbor v; lanes 16-31 -> point1 neighbor v
    const int  pme   = (lane < 16) ? p0 : p1;
    const bool valid = (lane < 16) || p1ok;
    const int cell = (pts[pme * 3 + 0] * Gdim + pts[pme * 3 + 1]) * Gdim + pts[pme * 3 + 2];
    const float bk2v = bk2[0];

#pragma unroll
    for (int n = 0; n < 4; ++n) {
        const int ch = n * 16 + nlane;
        const float b2v = b2[ch];
        float f[8];
#pragma unroll
        for (int k = 0; k < 8; ++k) f[k] = acc[n][k] + b2v;
        float f2v = bk2v;
#pragma unroll
        for (int o = 0; o < 8; ++o) {
            float s = bk1[o];
#pragma unroll
            for (int k = 0; k < 8; ++k) s += Wk1[o * 8 + k] * f[k];
            f2v += Wk2[o] * lrelu(s);
        }
        if (valid) atomicAdd(&sums[(size_t)cell * 64 + ch], f2v);
    }
    if (valid && nlane == 0) atomicAdd(&cnt[cell], 1.0f);  // lane 0 / lane 16
}

// ---------------------- 3x3x3 conv + ReLU (WMMA) ---------------------------
// Block = 4 waves -> (d, h0..h0+3). Wave = whole w-row (4 tiles) x 64 cout.
// Packed weights for the active kd-plane staged in LDS (144 KB), shared
// by all 4 waves; B reads become conflict-free ds_load_b64.
__global__ void __launch_bounds__(128)
conv3d_relu_wmma(const float* __restrict__ in,     // [G][G][G][64] (d,h,w,c)
                 const float* __restrict__ wPack,  // [27][16][4][64] packed B
                 const float* __restrict__ bias,   // [64]
                 float* __restrict__ out)          // [G][G][G][64]
{
    extern __shared__ float lds_w[];               // PLANE_PACK floats (144 KB)

    const int lane = threadIdx.x & 31;
    const int wv   = threadIdx.x >> 5;             // 0..3
    const int d    = blockIdx.x >> 4;              // block-uniform
    const int h    = ((blockIdx.x & 15) << 2) + wv;

    const int m    = lane & 15;                    // A row = w pos within tile
    const int ksub = (lane >> 4) << 1;             // 0 or 2

    v8f acc[4][4] = {};                            // [wtile][ntile]

    for (int kd = 0; kd < 3; ++kd) {               // block-uniform trip
        const int dd = d + kd - 1;
        if (dd < 0 || dd >= Gdim) continue;        // block-uniform skip

        // cooperative stage of this kd-plane's 9 packed taps into LDS
        {
            const float4* src = reinterpret_cast<const float4*>(wPack + kd * PLANE_PACK);
            float4* dst = reinterpret_cast<float4*>(lds_w);
            for (int i = threadIdx.x; i < PLANE_PACK / 4; i += 128) dst[i] = src[i];
        }
        __syncthreads();

        for (int kh = 0; kh < 3; ++kh) {           // wave-uniform
            const int hh = h + kh - 1;
            if (hh < 0 || hh >= Gdim) continue;
            const float* arow = in + (size_t)(dd * Gdim + hh) * Gdim * 64;
#pragma unroll
            for (int kw = 0; kw < 3; ++kw) {
                const float* lw = lds_w + (kh * 3 + kw) * TAP_PACK + lane * 2;
                const int wofs = m + kw - 1;
#pragma unroll 4
                for (int t = 0; t < 16; ++t) {
                    // B fragments from LDS (one b64 each, bank-conflict-free)
                    v2f b0 = *reinterpret_cast<const v2f*>(lw + (t * 4 + 0) * 64);
                    v2f b1 = *reinterpret_cast<const v2f*>(lw + (t * 4 + 1) * 64);
                    v2f b2 = *reinterpret_cast<const v2f*>(lw + (t * 4 + 2) * 64);
                    v2f b3 = *reinterpret_cast<const v2f*>(lw + (t * 4 + 3) * 64);
#pragma unroll
                    for (int wt = 0; wt < 4; ++wt) {
                        const int ww = wt * 16 + wofs;
                        v2f a = {0.0f, 0.0f};
                        if (ww >= 0 && ww < Gdim)
                            a = *reinterpret_cast<const v2f*>(
                                    arow + (size_t)ww * 64 + t * 4 + ksub);
                        acc[wt][0] = __builtin_amdgcn_wmma_f32_16x16x4_f32(
                            false, a, false, b0, (short)0, acc[wt][0], false, false);
                        acc[wt][1] = __builtin_amdgcn_wmma_f32_16x16x4_f32(
                            false, a, false, b1, (short)0, acc[wt][1], false, false);
                        acc[wt][2] = __builtin_amdgcn_wmma_f32_16x16x4_f32(
                            false, a, false, b2, (short)0, acc[wt][2], false, false);
                        acc[wt][3] = __builtin_amdgcn_wmma_f32_16x16x4_f32(
                            false, a, false, b3, (short)0, acc[wt][3], false, false);
                    }
                }
            }
        }
        __syncthreads();                           // protect LDS before restage
    }

    // epilogue: bias + relu + store (row M = v + 8*(lane>=16))
    const int nlane = lane & 15;
    const int moff  = (lane < 16) ? 0 : 8;
    const size_t sbase = (size_t)(d * Gdim + h) * Gdim;
#pragma unroll
    for (int wt = 0; wt < 4; ++wt) {
#pragma unroll
        for (int n = 0; n < 4; ++n) {
            const int ch = n * 16 + nlane;
            const float bv = bias[ch];
#pragma unroll
            for (int v = 0; v < 8; ++v) {
                float val = acc[wt][n][v] + bv;
                val = val > 0.0f ? val : 0.0f;
                out[(sbase + wt * 16 + v + moff) * 64 + ch] = val;
            }
        }
    }
}

// ------------------------------ launcher -----------------------------------

extern "C" void kernel_launch(void* const* d_in, const int* in_sizes, int n_in,
                              void* d_out, int out_size, void* d_ws, size_t ws_size,
                              hipStream_t stream) {
    const float* pc  = (const float*)d_in[0];
    const int*   pts = (const int*)  d_in[1];
    const float* W1  = (const float*)d_in[2];
    const float* b1  = (const float*)d_in[3];
    const float* W2  = (const float*)d_in[4];
    const float* b2  = (const float*)d_in[5];
    const float* Wk1 = (const float*)d_in[6];
    const float* bk1 = (const float*)d_in[7];
    const float* Wk2 = (const float*)d_in[8];
    const float* bk2 = (const float*)d_in[9];
    const float* Cw1 = (const float*)d_in[10];
    const float* Cb1 = (const float*)d_in[11];
    const float* Cw2 = (const float*)d_in[12];
    const float* Cb2 = (const float*)d_in[13];

    char* ws = (char*)d_ws;
    const size_t GRID_BYTES = (size_t)Gdim * Gdim * Gdim * Lch * sizeof(float); // 64 MB
    const size_t CNT_BYTES  = (size_t)Gdim * Gdim * Gdim * sizeof(float);       // 1 MB
    const size_t WP_BYTES   = (size_t)CONV_PACK * sizeof(float);                // 432 KB
    float* gridA  = (float*)(ws);                       // sums -> grid -> conv1 in
    float* gridB  = (float*)(ws + GRID_BYTES);          // conv1 out
    float* cnt    = (float*)(ws + 2 * GRID_BYTES);
    float* wPack1 = (float*)(ws + 2 * GRID_BYTES + CNT_BYTES);
    float* wPack2 = (float*)(ws + 2 * GRID_BYTES + CNT_BYTES + WP_BYTES);
    float* w2Pack = (float*)(ws + 2 * GRID_BYTES + CNT_BYTES + 2 * WP_BYTES);

    {
        int n = Gdim * Gdim * Gdim * Lch;
        zero_f32<<<(n + 255) / 256, 256, 0, stream>>>(gridA, n);
    }
    {
        int n = Gdim * Gdim * Gdim;
        zero_f32<<<(n + 255) / 256, 256, 0, stream>>>(cnt, n);
    }
    {
        int n = CONV_PACK;
        pack_conv_w<<<(n + 255) / 256, 256, 0, stream>>>(Cw1, wPack1);
        pack_conv_w<<<(n + 255) / 256, 256, 0, stream>>>(Cw2, wPack2);
        pack_w2<<<(TAP_PACK + 255) / 256, 256, 0, stream>>>(W2, w2Pack);
    }
    {
        int waves  = (NPTS + 1) / 2;
        int blocks = (waves + 3) / 4;   // 4 waves / block
        point_encode_scatter<<<blocks, 128, 0, stream>>>(
            pc, pts, W1, b1, w2Pack, b2, Wk1, bk1, Wk2, bk2, gridA, cnt);
    }
    {
        int n = Gdim * Gdim * Gdim * Lch;
        finalize_grid<<<(n + 255) / 256, 256, 0, stream>>>(gridA, cnt);
    }
    const size_t conv_lds = (size_t)PLANE_PACK * sizeof(float);  // 144 KB
    conv3d_relu_wmma<<<Gdim * 16, 128, conv_lds, stream>>>(gridA, wPack1, Cb1, gridB);
    conv3d_relu_wmma<<<Gdim * 16, 128, conv_lds, stream>>>(gridB, wPack2, Cb2, (float*)d_out);
}